// DistributedGCN_82703890251955
// MI455X (gfx1250) — compile-verified
//
#include <hip/hip_runtime.h>

#define DCH 64   // feature dim

typedef float v2f __attribute__((ext_vector_type(2)));
typedef float v8f __attribute__((ext_vector_type(8)));

__device__ __forceinline__ void atomAddF(float* p, float v) {
    // lowers to global_atomic_add_f32 on gfx1250
    unsafeAtomicAdd(p, v);
}

// ----------------------- degree / symmetric norm -----------------------
__global__ __launch_bounds__(256) void k_deg_init(float* __restrict__ deg, int n) {
    int i = blockIdx.x * blockDim.x + threadIdx.x;
    if (i < n) deg[i] = 1.0f;               // self-loop contribution
}

__global__ __launch_bounds__(256) void k_deg_count(const int* __restrict__ dst,
                                                   float* __restrict__ deg, int e) {
    int i = blockIdx.x * blockDim.x + threadIdx.x;
    if (i < e) atomAddF(&deg[dst[i]], 1.0f);
}

__global__ __launch_bounds__(256) void k_dinv(float* __restrict__ deg, int n) {
    int i = blockIdx.x * blockDim.x + threadIdx.x;
    if (i < n) deg[i] = rsqrtf(deg[i]);     // deg >= 1 always (self loops)
}

// ----------------------- fp32 WMMA GEMM: H = X (n x 64) @ W (64 x 64) ---
// One wave computes a 16-row strip: 4 tiles of 16 cols, K-loop of 16 x4 steps.
__global__ __launch_bounds__(256) void k_gemm_wmma(const float* __restrict__ X,
                                                   const float* __restrict__ W,
                                                   float* __restrict__ H, int nrows) {
    const int lane  = threadIdx.x & 31;
    const int wave  = threadIdx.x >> 5;
    const int m0    = (blockIdx.x * 8 + wave) * 16;   // wave-uniform guard
    if (m0 >= nrows) return;                          // EXEC stays all-1s
    const int ml    = lane & 15;
    const int khalf = lane >> 4;                      // lanes16-31 hold K+2

    // A fragments: row m0+ml, 16 k-steps of 4 (ISA 32-bit 16x4 A layout)
    v2f a[16];
    const float* __restrict__ xrow = X + (size_t)(m0 + ml) * DCH;
    #pragma unroll
    for (int kk = 0; kk < 16; ++kk) {
        const int kb = kk * 4 + khalf * 2;
        a[kk].x = xrow[kb];
        a[kk].y = xrow[kb + 1];
    }

    #pragma unroll
    for (int nt = 0; nt < 4; ++nt) {
        const int n0 = nt * 16;
        v8f c = {};
        #pragma unroll
        for (int kk = 0; kk < 16; ++kk) {
            const int kb = kk * 4 + khalf * 2;
            v2f b;
            b.x = W[(size_t)kb * DCH + n0 + ml];
            b.y = W[(size_t)(kb + 1) * DCH + n0 + ml];
            c = __builtin_amdgcn_wmma_f32_16x16x4_f32(
                    false, a[kk], false, b, (short)0, c, false, false);
        }
        #pragma unroll
        for (int r = 0; r < 8; ++r) {
            const int row = m0 + r + khalf * 8;
            H[(size_t)row * DCH + n0 + ml] = c[r];
        }
    }
}

// ----------------------- out = bias + h * dinv^2 (self loop + bias) -----
__global__ __launch_bounds__(256) void k_self_init(const float* __restrict__ H,
                                                   const float* __restrict__ dinv,
                                                   const float* __restrict__ bias,
                                                   float* __restrict__ out, int n16) {
    int t = blockIdx.x * blockDim.x + threadIdx.x;   // (row, 4-col chunk)
    if (t >= n16) return;
    const int row = t >> 4;
    const int c   = (t & 15) * 4;
    float w = dinv[row]; w = w * w;
    const float4 h = *(const float4*)(H + (size_t)t * 4);
    float4 o;
    o.x = bias[c + 0] + h.x * w;
    o.y = bias[c + 1] + h.y * w;
    o.z = bias[c + 2] + h.z * w;
    o.w = bias[c + 3] + h.w * w;
    *(float4*)(out + (size_t)t * 4) = o;
}

// ----------------------- edge scatter-add: 16 threads per edge ----------
__global__ __launch_bounds__(256) void k_edge_agg(const float* __restrict__ H,
                                                  const int* __restrict__ src,
                                                  const int* __restrict__ dst,
                                                  const float* __restrict__ dinv,
                                                  float* __restrict__ out, int e) {
    int t = blockIdx.x * blockDim.x + threadIdx.x;
    int ed = t >> 4;
    if (ed >= e) return;
    const int c = (t & 15) * 4;
    const int s = src[ed];
    const int d = dst[ed];
    const float w = dinv[s] * dinv[d];
    const float4 h = *(const float4*)(H + (size_t)s * DCH + c);
    float* o = out + (size_t)d * DCH + c;
    atomAddF(o + 0, h.x * w);
    atomAddF(o + 1, h.y * w);
    atomAddF(o + 2, h.z * w);
    atomAddF(o + 3, h.w * w);
}

// ----------------------- batch norm (training) + relu ------------------
__global__ void k_zero_stats(float* __restrict__ stats) {
    if (threadIdx.x < 128) stats[threadIdx.x] = 0.0f;
}

__global__ __launch_bounds__(256) void k_bn_reduce(const float* __restrict__ X,
                                                   float* __restrict__ stats, int n) {
    __shared__ float ssum[256];
    __shared__ float ssq[256];
    const int col = threadIdx.x & 63;
    const int grp = threadIdx.x >> 6;                 // 0..3
    float s = 0.f, q = 0.f;
    for (int r = blockIdx.x * 4 + grp; r < n; r += gridDim.x * 4) {
        const float v = X[(size_t)r * DCH + col];
        s += v; q += v * v;
    }
    ssum[threadIdx.x] = s; ssq[threadIdx.x] = q;
    __syncthreads();
    if (threadIdx.x < 64) {
        float ts = ssum[threadIdx.x] + ssum[threadIdx.x + 64] +
                   ssum[threadIdx.x + 128] + ssum[threadIdx.x + 192];
        float tq = ssq[threadIdx.x] + ssq[threadIdx.x + 64] +
                   ssq[threadIdx.x + 128] + ssq[threadIdx.x + 192];
        atomAddF(&stats[threadIdx.x], ts);
        atomAddF(&stats[64 + threadIdx.x], tq);
    }
}

__global__ void k_bn_finalize(float* __restrict__ stats, const float* __restrict__ g,
                              const float* __restrict__ be, float n) {
    const int c = threadIdx.x;
    if (c < 64) {
        const float mean = stats[c] / n;
        const float var  = stats[64 + c] / n - mean * mean;   // biased var
        const float a    = g[c] * rsqrtf(var + 1e-5f);
        stats[128 + c] = a;
        stats[192 + c] = be[c] - mean * a;
    }
}

__global__ __launch_bounds__(256) void k_bn_apply_relu(float* __restrict__ X,
                                                       const float* __restrict__ stats,
                                                       int n16) {
    int t = blockIdx.x * blockDim.x + threadIdx.x;
    if (t >= n16) return;
    const int c = (t & 15) * 4;
    float4 v = *(float4*)(X + (size_t)t * 4);
    v.x = fmaxf(v.x * stats[128 + c + 0] + stats[192 + c + 0], 0.f);
    v.y = fmaxf(v.y * stats[128 + c + 1] + stats[192 + c + 1], 0.f);
    v.z = fmaxf(v.z * stats[128 + c + 2] + stats[192 + c + 2], 0.f);
    v.w = fmaxf(v.w * stats[128 + c + 3] + stats[192 + c + 3], 0.f);
    *(float4*)(X + (size_t)t * 4) = v;
}

// ----------------------- launcher ---------------------------------------
extern "C" void kernel_launch(void* const* d_in, const int* in_sizes, int n_in,
                              void* d_out, int out_size, void* d_ws, size_t ws_size,
                              hipStream_t stream) {
    const float* x   = (const float*)d_in[0];
    const int*   ei  = (const int*)d_in[1];       // (2, E) flattened
    const float* W1  = (const float*)d_in[2];
    const float* b1  = (const float*)d_in[3];
    const float* g1  = (const float*)d_in[4];
    const float* be1 = (const float*)d_in[5];
    const float* W2  = (const float*)d_in[6];
    const float* b2  = (const float*)d_in[7];
    const float* g2  = (const float*)d_in[8];
    const float* be2 = (const float*)d_in[9];
    const float* W3  = (const float*)d_in[10];
    const float* b3  = (const float*)d_in[11];

    const int n = in_sizes[0] / DCH;              // 100000
    const int e = in_sizes[1] / 2;                // 1200000
    const int* esrc = ei;
    const int* edst = ei + e;

    float* F     = (float*)d_ws;
    float* dinv  = F;                              // n floats
    float* stats = F + n;                          // 256 floats
    float* h     = stats + 256;                    // n*64
    float* buf   = h + (size_t)n * DCH;            // n*64

    const int n16       = n * 16;
    const int nodeBlks  = (n + 255) / 256;
    const int n16Blks   = (n16 + 255) / 256;
    const int edgeBlks  = (e + 255) / 256;
    const int aggBlks   = (e * 16 + 255) / 256;
    const int gemmBlks  = (n / 16 + 7) / 8;        // 8 waves per block

    // symmetric normalization
    k_deg_init <<<nodeBlks, 256, 0, stream>>>(dinv, n);
    k_deg_count<<<edgeBlks, 256, 0, stream>>>(edst, dinv, e);
    k_dinv     <<<nodeBlks, 256, 0, stream>>>(dinv, n);

    // ---- layer 1 ----
    k_gemm_wmma<<<gemmBlks, 256, 0, stream>>>(x, W1, h, n);
    k_self_init<<<n16Blks, 256, 0, stream>>>(h, dinv, b1, buf, n16);
    k_edge_agg <<<aggBlks, 256, 0, stream>>>(h, esrc, edst, dinv, buf, e);
    k_zero_stats<<<1, 128, 0, stream>>>(stats);
    k_bn_reduce <<<512, 256, 0, stream>>>(buf, stats, n);
    k_bn_finalize<<<1, 64, 0, stream>>>(stats, g1, be1, (float)n);
    k_bn_apply_relu<<<n16Blks, 256, 0, stream>>>(buf, stats, n16);

    // ---- layer 2 ----
    k_gemm_wmma<<<gemmBlks, 256, 0, stream>>>(buf, W2, h, n);
    k_self_init<<<n16Blks, 256, 0, stream>>>(h, dinv, b2, buf, n16);
    k_edge_agg <<<aggBlks, 256, 0, stream>>>(h, esrc, edst, dinv, buf, e);
    k_zero_stats<<<1, 128, 0, stream>>>(stats);
    k_bn_reduce <<<512, 256, 0, stream>>>(buf, stats, n);
    k_bn_finalize<<<1, 64, 0, stream>>>(stats, g2, be2, (float)n);
    k_bn_apply_relu<<<n16Blks, 256, 0, stream>>>(buf, stats, n16);

    // ---- layer 3 (output straight to d_out) ----
    float* out = (float*)d_out;
    k_gemm_wmma<<<gemmBlks, 256, 0, stream>>>(buf, W3, h, n);
    k_self_init<<<n16Blks, 256, 0, stream>>>(h, dinv, b3, out, n16);
    k_edge_agg <<<aggBlks, 256, 0, stream>>>(h, esrc, edst, dinv, out, e);
}